// SAGE_78451872629304
// MI455X (gfx1250) — compile-verified
//
#include <hip/hip_runtime.h>

#define NODE_F 128   // IN_F == HID_F == OUT_F == 128

typedef float v2f __attribute__((ext_vector_type(2)));
typedef float v8f __attribute__((ext_vector_type(8)));

// D = A(16x4) * B(4x16) + C, fp32 WMMA. NEG for f32 is C-neg only -> pass false.
static __device__ __forceinline__ v8f wmma_k4(v2f a, v2f b, v8f c) {
  return __builtin_amdgcn_wmma_f32_16x16x4_f32(
      /*neg_a=*/false, a, /*neg_b=*/false, b,
      /*c_mod=*/(short)0, c, /*reuse_a=*/false, /*reuse_b=*/false);
}

// ---------------- degree: one thread per edge ----------------
__global__ void deg_kernel(const int* __restrict__ dst, float* __restrict__ deg, int nEdges) {
  int e = blockIdx.x * blockDim.x + threadIdx.x;
  if (e < nEdges) atomicAdd(&deg[dst[e]], 1.0f);
}

// ---------------- scatter-add: one wave (32 lanes) per edge ----------------
// Each lane moves one float4 => 128 floats per edge, coalesced on both sides.
__global__ void scatter_kernel(const float* __restrict__ h, const int* __restrict__ src,
                               const int* __restrict__ dst, float* __restrict__ sum,
                               int nEdges) {
  int g    = blockIdx.x * blockDim.x + threadIdx.x;
  int e    = g >> 5;
  int lane = g & 31;
  if (e >= nEdges) return;
  int s = src[e];
  int d = dst[e];
  float4 v = ((const float4*)(h + (size_t)s * NODE_F))[lane];
  float* sp = sum + (size_t)d * NODE_F + lane * 4;
  atomicAdd(sp + 0, v.x);
  atomicAdd(sp + 1, v.y);
  atomicAdd(sp + 2, v.z);
  atomicAdd(sp + 3, v.w);
}

// ---------------- fused SAGE GEMM ----------------
// out[m,:] = hself[m,:] @ Wself^T (+ (sum[m,:]/max(deg,1)) @ Wneigh^T) + bias, opt ReLU.
// Block: 256 threads (8 waves), computes 32 rows x 128 cols.
// Wave w owns N-tile n0 = 16*w; holds BOTH 16-row accumulators live so each
// weight (B) load feeds two V_WMMA_F32_16X16X4_F32, and consecutive WMMAs hit
// independent accumulators (better XDL pipelining than a single RAW chain).
template <bool NEIGH, bool RELU>
__global__ void sage_gemm(const float* __restrict__ hself, const float* __restrict__ sum,
                          const float* __restrict__ deg, const float* __restrict__ Wself,
                          const float* __restrict__ Wneigh, const float* __restrict__ bias,
                          float* __restrict__ out, int nNodes) {
  __shared__ float hs[32][NODE_F + 4];
  __shared__ float hn[32][NODE_F + 4];

  const int tid     = threadIdx.x;
  const int m_base  = blockIdx.x * 32;
  const bool fullTile = (m_base + 32 <= nNodes);   // uniform per block

  // ---- stage 32 rows of hself (and mean-normalized neighbor sums) into LDS ----
  // 32*128 floats = 1024 float4's; 256 threads x 4 iters.
  for (int i = 0; i < 4; ++i) {
    int f   = i * 256 + tid;   // float4 index
    int row = f >> 5;          // 0..31
    int cv  = f & 31;          // float4 column
    int m   = m_base + row;
    float4 v = make_float4(0.f, 0.f, 0.f, 0.f);
    if (fullTile || m < nNodes) v = ((const float4*)(hself + (size_t)m * NODE_F))[cv];
    *(float4*)&hs[row][cv * 4] = v;
    if (NEIGH) {
      float4 s4 = make_float4(0.f, 0.f, 0.f, 0.f);
      float rd = 1.0f;
      if (fullTile || m < nNodes) {
        s4 = ((const float4*)(sum + (size_t)m * NODE_F))[cv];
        rd = 1.0f / fmaxf(deg[m], 1.0f);
      }
      hn[row][cv * 4 + 0] = s4.x * rd;
      hn[row][cv * 4 + 1] = s4.y * rd;
      hn[row][cv * 4 + 2] = s4.z * rd;
      hn[row][cv * 4 + 3] = s4.w * rd;
    }
  }
  __syncthreads();

  const int lane = tid & 31;
  const int wv   = tid >> 5;
  const int lid  = lane & 15;   // lane within half
  const int half = lane >> 4;   // 0: K pair {0,1} / rows 0-7 ; 1: K pair {2,3} / rows 8-15
  const int n0   = wv * 16;     // this wave's output column tile

  const float* Ws = Wself + (size_t)(n0 + lid) * NODE_F;
  const float* Wn = NEIGH ? (Wneigh + (size_t)(n0 + lid) * NODE_F) : nullptr;

  v8f acc0 = {0.f, 0.f, 0.f, 0.f, 0.f, 0.f, 0.f, 0.f};   // rows m_base+0..15
  v8f acc1 = {0.f, 0.f, 0.f, 0.f, 0.f, 0.f, 0.f, 0.f};   // rows m_base+16..31

#pragma unroll 8
  for (int k0 = 0; k0 < NODE_F; k0 += 4) {
    const int kk = k0 + 2 * half;
    // B: 4x16 tile = Wself[n0+lid][kk..kk+1] (B = W-tile transposed for h @ W^T)
    v2f bs;
    bs.x = Ws[kk];
    bs.y = Ws[kk + 1];
    // A: 16x4 fp32 tiles (ISA layout: lanes 0-15 hold K=0,1; lanes 16-31 K=2,3)
    v2f a0, a1;
    a0.x = hs[lid][kk];
    a0.y = hs[lid][kk + 1];
    a1.x = hs[16 + lid][kk];
    a1.y = hs[16 + lid][kk + 1];
    acc0 = wmma_k4(a0, bs, acc0);
    acc1 = wmma_k4(a1, bs, acc1);
    if (NEIGH) {
      v2f bn;
      bn.x = Wn[kk];
      bn.y = Wn[kk + 1];
      v2f c0, c1;
      c0.x = hn[lid][kk];
      c0.y = hn[lid][kk + 1];
      c1.x = hn[16 + lid][kk];
      c1.y = hn[16 + lid][kk + 1];
      acc0 = wmma_k4(c0, bn, acc0);
      acc1 = wmma_k4(c1, bn, acc1);
    }
  }

  // ---- epilogue: C/D layout: VGPR e -> row M = e + 8*half, col N = lane%16 ----
  const float bv = bias[n0 + lid];
  float* outc = out + n0 + lid;
  const int r0 = m_base + 8 * half;        // acc0 row base
  const int r1 = m_base + 16 + 8 * half;   // acc1 row base

  if (fullTile) {
#pragma unroll
    for (int e = 0; e < 8; ++e) {
      float v0 = acc0[e] + bv;
      float v1 = acc1[e] + bv;
      if (RELU) { v0 = fmaxf(v0, 0.f); v1 = fmaxf(v1, 0.f); }
      outc[(size_t)(r0 + e) * NODE_F] = v0;
      outc[(size_t)(r1 + e) * NODE_F] = v1;
    }
  } else {
#pragma unroll
    for (int e = 0; e < 8; ++e) {
      float v0 = acc0[e] + bv;
      float v1 = acc1[e] + bv;
      if (RELU) { v0 = fmaxf(v0, 0.f); v1 = fmaxf(v1, 0.f); }
      if (r0 + e < nNodes) outc[(size_t)(r0 + e) * NODE_F] = v0;
      if (r1 + e < nNodes) outc[(size_t)(r1 + e) * NODE_F] = v1;
    }
  }
}

extern "C" void kernel_launch(void* const* d_in, const int* in_sizes, int n_in,
                              void* d_out, int out_size, void* d_ws, size_t ws_size,
                              hipStream_t stream) {
  const float* x      = (const float*)d_in[0];
  const int*   src    = (const int*)d_in[1];
  const int*   dst    = (const int*)d_in[2];
  const float* W1s    = (const float*)d_in[3];
  const float* W1n    = (const float*)d_in[4];
  const float* b1     = (const float*)d_in[5];
  const float* W2s    = (const float*)d_in[6];
  const float* W2n    = (const float*)d_in[7];
  const float* b2     = (const float*)d_in[8];
  const float* W3     = (const float*)d_in[9];
  const float* b3     = (const float*)d_in[10];
  float*       out    = (float*)d_out;

  const int nNodes = in_sizes[0] / NODE_F;   // 100000
  const int nEdges = in_sizes[1];            // 1600000

  // scratch layout (floats): sum[N*128] | deg[N] | h1[N*128] | h2[N*128]
  float* sum = (float*)d_ws;
  float* deg = sum + (size_t)nNodes * NODE_F;
  float* h1  = deg + nNodes;
  float* h2  = h1 + (size_t)nNodes * NODE_F;

  dim3 blk(256);
  dim3 gEdge((nEdges + 255) / 256);
  dim3 gScat((nEdges * 32 + 255) / 256);
  dim3 gGemm((nNodes + 31) / 32);

  // ----- layer 1 -----
  hipMemsetAsync(sum, 0, ((size_t)nNodes * NODE_F + nNodes) * sizeof(float), stream); // sum+deg
  deg_kernel<<<gEdge, blk, 0, stream>>>(dst, deg, nEdges);
  scatter_kernel<<<gScat, blk, 0, stream>>>(x, src, dst, sum, nEdges);
  sage_gemm<true, true><<<gGemm, blk, 0, stream>>>(x, sum, deg, W1s, W1n, b1, h1, nNodes);

  // ----- layer 2 -----
  hipMemsetAsync(sum, 0, (size_t)nNodes * NODE_F * sizeof(float), stream);
  scatter_kernel<<<gScat, blk, 0, stream>>>(h1, src, dst, sum, nEdges);
  sage_gemm<true, false><<<gGemm, blk, 0, stream>>>(h1, sum, deg, W2s, W2n, b2, h2, nNodes);

  // ----- final linear -----
  sage_gemm<false, false><<<gGemm, blk, 0, stream>>>(h2, nullptr, nullptr, W3, nullptr, b3,
                                                     out, nNodes);
}